// SmoothGCN_19155554140398
// MI455X (gfx1250) — compile-verified
//
#include <hip/hip_runtime.h>

#define N_NODES 100000
#define N_EDGES 800000
#define F_IN    256
#define HID     128
#define N_CLS   32

typedef __attribute__((ext_vector_type(2))) float v2f;
typedef __attribute__((ext_vector_type(8))) float v8f;

// ---------------------------------------------------------------------------
// degree / normalization
// ---------------------------------------------------------------------------
__global__ void init_deg_kernel(float* __restrict__ deg, int n) {
    int i = blockIdx.x * blockDim.x + threadIdx.x;
    if (i < n) deg[i] = 1.0f;  // self-loop contribution
}

__global__ void deg_scatter_kernel(const long long* __restrict__ dst,
                                   float* __restrict__ deg, int ne) {
    int e = blockIdx.x * blockDim.x + threadIdx.x;
    if (e < ne) {
        int d = (int)__builtin_nontemporal_load(&dst[e]);  // streaming read
        atomicAdd(&deg[d], 1.0f);
    }
}

__global__ void dinv_kernel(const float* __restrict__ deg,
                            float* __restrict__ dinv, int n) {
    int i = blockIdx.x * blockDim.x + threadIdx.x;
    if (i < n) {
        float d = deg[i];
        dinv[i] = d > 0.0f ? rsqrtf(d) : 0.0f;
    }
}

// ---------------------------------------------------------------------------
// dense GEMM  Y[n_nodes, NT*16] = X[n_nodes, K] @ W[K, NT*16]
// fp32 WMMA 16x16x4; one wave owns a 16-node strip and all NT output tiles.
// A 16x4 layout : lane m=L&15 is row M; VGPR0 holds K= {0 | 2}, VGPR1 K={1|3}
//                 split by lane half (kh = L>>4).
// B 4x16 layout : lane m is col N, same K split per lane half.
// C/D 16x16     : VGPR v -> row (v + 8*kh), col m.
// X is streamed once -> non-temporal loads keep L2 free for the downstream
// gather/scatter working set (xw + h, ~100 MB, fits the 192 MB L2).
// ---------------------------------------------------------------------------
template <int K, int NT, bool XNT>
__global__ __launch_bounds__(128) void gemm_wmma_kernel(
    const float* __restrict__ X, const float* __restrict__ W,
    float* __restrict__ Y, int n_nodes) {
    constexpr int NOUT = NT * 16;
    const int lane  = threadIdx.x & 31;
    const int wid   = threadIdx.x >> 5;
    const int node0 = (blockIdx.x * 4 + wid) * 16;
    if (node0 >= n_nodes) return;  // wave-uniform: EXEC stays all-ones for WMMA

    const int m  = lane & 15;   // row of A / col of B / col of C
    const int kh = lane >> 4;   // K-pair select (0: K+0/1, 1: K+2/3)

    v8f acc[NT];
    v8f zero = {};
#pragma unroll
    for (int t = 0; t < NT; ++t) acc[t] = zero;

    const float* xrow = X + (long)(node0 + m) * K;
    for (int kk = 0; kk < K; kk += 4) {
        const int ka = kk + 2 * kh;
        v2f a;
        if (XNT) {
            a.x = __builtin_nontemporal_load(&xrow[ka]);
            a.y = __builtin_nontemporal_load(&xrow[ka + 1]);
        } else {
            a.x = xrow[ka];
            a.y = xrow[ka + 1];
        }
#pragma unroll
        for (int t = 0; t < NT; ++t) {
            const int n = t * 16 + m;
            v2f b;
            b.x = W[(long)ka * NOUT + n];
            b.y = W[(long)(ka + 1) * NOUT + n];
            acc[t] = __builtin_amdgcn_wmma_f32_16x16x4_f32(
                false, a, false, b, (short)0, acc[t], false, false);
        }
    }

#pragma unroll
    for (int t = 0; t < NT; ++t)
#pragma unroll
        for (int v = 0; v < 8; ++v)
            Y[(long)(node0 + v + 8 * kh) * NOUT + t * 16 + m] = acc[t][v];
}

// ---------------------------------------------------------------------------
// H[i] = XW[i] * dinv[node]^2   (self-loop term; fully initializes H)
// ---------------------------------------------------------------------------
template <int F>
__global__ void self_loop_init_kernel(const float* __restrict__ XW,
                                      const float* __restrict__ dinv,
                                      float* __restrict__ H, int n) {
    int i = blockIdx.x * blockDim.x + threadIdx.x;
    if (i >= n * F) return;
    int node = i / F;
    float di = dinv[node];
    H[i] = XW[i] * di * di;
}

// ---------------------------------------------------------------------------
// edge scatter: H[dst] += XW[src] * dinv[src]*dinv[dst]
// edge list is a pure stream (NT); XW gathers and H atomics want L2 residency.
// ---------------------------------------------------------------------------
template <int F>
__global__ void edge_scatter_kernel(const long long* __restrict__ src,
                                    const long long* __restrict__ dst,
                                    const float* __restrict__ dinv,
                                    const float* __restrict__ XW,
                                    float* __restrict__ H, int ne) {
    constexpr int EPB = 256 / F;
    int e = blockIdx.x * EPB + (int)(threadIdx.x / F);
    int f = threadIdx.x % F;
    if (e >= ne) return;
    int s = (int)__builtin_nontemporal_load(&src[e]);
    int d = (int)__builtin_nontemporal_load(&dst[e]);
    float nrm = dinv[s] * dinv[d];
    atomicAdd(&H[(long)d * F + f], XW[(long)s * F + f] * nrm);
}

// ---------------------------------------------------------------------------
// H = relu(H + b)
// ---------------------------------------------------------------------------
__global__ void bias_relu_kernel(float* __restrict__ H,
                                 const float* __restrict__ b,
                                 int total, int F) {
    int i = blockIdx.x * blockDim.x + threadIdx.x;
    if (i < total) H[i] = fmaxf(H[i] + b[i % F], 0.0f);
}

// ---------------------------------------------------------------------------
// out = log_softmax(H + b), 32 classes: one wave32 per node, lane = class.
// ---------------------------------------------------------------------------
__global__ void logsoftmax32_kernel(const float* __restrict__ H,
                                    const float* __restrict__ b,
                                    float* __restrict__ out, int n) {
    int wave = (blockIdx.x * blockDim.x + threadIdx.x) >> 5;
    int lane = threadIdx.x & 31;
    if (wave >= n) return;
    float z = H[(long)wave * 32 + lane] + b[lane];
    float mx = z;
#pragma unroll
    for (int o = 16; o > 0; o >>= 1) mx = fmaxf(mx, __shfl_xor(mx, o, 32));
    float ex = expf(z - mx);
    float sum = ex;
#pragma unroll
    for (int o = 16; o > 0; o >>= 1) sum += __shfl_xor(sum, o, 32);
    // result is only read back by the host -> non-temporal store
    __builtin_nontemporal_store(z - mx - logf(sum), &out[(long)wave * 32 + lane]);
}

// ---------------------------------------------------------------------------
extern "C" void kernel_launch(void* const* d_in, const int* in_sizes, int n_in,
                              void* d_out, int out_size, void* d_ws, size_t ws_size,
                              hipStream_t stream) {
    const float*     X   = (const float*)d_in[0];
    const long long* ei  = (const long long*)d_in[1];
    const float*     W1  = (const float*)d_in[2];
    const float*     b1  = (const float*)d_in[3];
    const float*     W2  = (const float*)d_in[4];
    const float*     b2  = (const float*)d_in[5];
    const long long* src = ei;
    const long long* dst = ei + N_EDGES;
    float*           out = (float*)d_out;

    // workspace layout (floats)
    float* ws   = (float*)d_ws;
    float* deg  = ws;                               // 100000 (+pad)
    float* dinv = deg  + 100352;                    // 100000 (+pad)
    float* xw1  = dinv + 100352;                    // 100000*128
    float* h1   = xw1  + (long)N_NODES * HID;       // 100000*128
    float* xw2  = h1   + (long)N_NODES * HID;       // 100000*32
    // layer-2 accumulation happens directly in d_out

    const int T = 256;

    // normalization
    init_deg_kernel<<<(N_NODES + T - 1) / T, T, 0, stream>>>(deg, N_NODES);
    deg_scatter_kernel<<<(N_EDGES + T - 1) / T, T, 0, stream>>>(dst, deg, N_EDGES);
    dinv_kernel<<<(N_NODES + T - 1) / T, T, 0, stream>>>(deg, dinv, N_NODES);

    const int node_tiles = N_NODES / 16;            // 6250 (exact)
    const int gemm_blocks = (node_tiles + 3) / 4;   // 4 waves / block

    // layer 1: xw1 = X @ W1  (fp32 WMMA, X streamed non-temporally)
    gemm_wmma_kernel<F_IN, HID / 16, true><<<gemm_blocks, 128, 0, stream>>>(X, W1, xw1, N_NODES);
    self_loop_init_kernel<HID><<<(N_NODES * HID + T - 1) / T, T, 0, stream>>>(xw1, dinv, h1, N_NODES);
    edge_scatter_kernel<HID><<<(N_EDGES + 1) / 2, T, 0, stream>>>(src, dst, dinv, xw1, h1, N_EDGES);
    bias_relu_kernel<<<(N_NODES * HID + T - 1) / T, T, 0, stream>>>(h1, b1, N_NODES * HID, HID);

    // layer 2: xw2 = h1 @ W2 (fp32 WMMA; h1 is L2-hot, keep RT hints)
    gemm_wmma_kernel<HID, N_CLS / 16, false><<<gemm_blocks, 128, 0, stream>>>(h1, W2, xw2, N_NODES);
    self_loop_init_kernel<N_CLS><<<(N_NODES * N_CLS + T - 1) / T, T, 0, stream>>>(xw2, dinv, out, N_NODES);
    edge_scatter_kernel<N_CLS><<<(N_EDGES + 7) / 8, T, 0, stream>>>(src, dst, dinv, xw2, out, N_EDGES);

    // bias + log_softmax in place on d_out
    logsoftmax32_kernel<<<(N_NODES + 7) / 8, T, 0, stream>>>(out, b2, out, N_NODES);
}